// MultiScaleBSQTP_88708254532113
// MI455X (gfx1250) — compile-verified
//
#include <hip/hip_runtime.h>
#include <math.h>

typedef __attribute__((ext_vector_type(2))) float v2f;
typedef __attribute__((ext_vector_type(8))) float v8f;

#define BB 8
#define CC 64
#define TDIM 5
#define HDIM 64
#define WDIM 64
#define NELEM (BB*CC*TDIM*HDIM*WDIM)      /* 10485760 */
#define ROWS_W (BB*CC*TDIM*HDIM)          /* 163840 rows for W-axis GEMMs */

// ---------------- init: residual = x, out accumulator = 0, zero reductions ----
__global__ void ms_init(const float* __restrict__ x, float* __restrict__ R,
                        float* __restrict__ outAcc, float* __restrict__ red,
                        int n, int nred) {
  int i = blockIdx.x * blockDim.x + threadIdx.x;
  if (i < n) { R[i] = x[i]; outAcc[i] = 0.0f; }
  if (i < nred) red[i] = 0.0f;
}

// ---------------- jax.image.resize 'trilinear' weight rows (triangle kernel,
// antialias on downsample, normalized) -------------------------------------
__global__ void ms_weights(float* __restrict__ Wm, int n_in, int n_out) {
  int i = threadIdx.x;
  if (i >= n_out) return;
  float inv_scale = (float)n_in / (float)n_out;
  float ks = inv_scale > 1.0f ? inv_scale : 1.0f;   // antialias kernel width
  float sf = ((float)i + 0.5f) * inv_scale - 0.5f;
  float total = 0.0f;
  for (int j = 0; j < n_in; ++j) {
    float xx = fabsf(sf - (float)j) / ks;
    total += (xx < 1.0f) ? (1.0f - xx) : 0.0f;
  }
  float invt = (fabsf(total) > 1.1920929e-4f) ? (1.0f / total) : 0.0f;
  for (int j = 0; j < n_in; ++j) {
    float xx = fabsf(sf - (float)j) / ks;
    float wgt = (xx < 1.0f) ? (1.0f - xx) : 0.0f;
    Wm[i * n_in + j] = wgt * invt;
  }
}

// ---------------- generic scalar axis resize: out[o,i,inr] = sum_j W[i,j] in[o,j,inr]
__global__ void ms_resize_axis(const float* __restrict__ in, float* __restrict__ out,
                               const float* __restrict__ Wt,
                               int outer, int n_in, int n_out, int inner) {
  int idx = blockIdx.x * blockDim.x + threadIdx.x;
  long total = (long)outer * n_out * inner;
  if (idx >= total) return;
  int inr = idx % inner;
  int t2  = idx / inner;
  int i   = t2 % n_out;
  int o   = t2 / n_out;
  const float* src  = in + (long)o * n_in * inner + inr;
  const float* wrow = Wt + (long)i * n_in;
  float acc = 0.0f;
  for (int j = 0; j < n_in; ++j)
    acc = fmaf(wrow[j], src[(long)j * inner], acc);
  out[idx] = acc;
}

// scalar fused final W-upsample: out += val, residual -= val
__global__ void ms_resize_axis_accum(const float* __restrict__ in,
                                     const float* __restrict__ Wt,
                                     float* __restrict__ outAcc, float* __restrict__ resid,
                                     int outer, int n_in, int n_out) {
  int idx = blockIdx.x * blockDim.x + threadIdx.x;
  long total = (long)outer * n_out;
  if (idx >= total) return;
  int i = idx % n_out;
  int o = idx / n_out;
  const float* src  = in + (long)o * n_in;
  const float* wrow = Wt + (long)i * n_in;
  float acc = 0.0f;
  for (int j = 0; j < n_in; ++j)
    acc = fmaf(wrow[j], src[j], acc);
  outAcc[idx] += acc;
  resid[idx]  -= acc;
}

// ---------------- WMMA f32 16x16x4 W-axis resize GEMM ----------------------
// rows = ROWS_W (divisible by 16), n_in,n_out multiples of 16 -> every wave has
// a full tile, EXEC is all ones as WMMA requires.
// A layout (16x4 f32): lane<16: M=lane, regs={K0,K1}; lane>=16: M=lane-16, regs={K2,K3}
// B layout (4x16 f32): reg v: lanes<16 K=v, lanes>=16 K=v+2; N = lane%16
// C/D layout (16x16 f32): reg v: M = v + 8*(lane/16); N = lane%16
__global__ void ms_resize_w_wmma(const float* __restrict__ in, float* __restrict__ out,
                                 const float* __restrict__ Wt, int n_in, int n_out) {
  int wave = (blockIdx.x * blockDim.x + threadIdx.x) >> 5;
  int lane = threadIdx.x & 31;
  int tilesN = n_out >> 4;
  int tileM = wave / tilesN;
  int tileN = wave - tileM * tilesN;
  int row0 = tileM << 4;
  int col0 = tileN << 4;
  int mrow  = lane & 15;
  int khalf = lane >> 4;
  const float* arow = in + (size_t)(row0 + mrow) * n_in + 2 * khalf;
  const float* brow = Wt + (size_t)(col0 + mrow) * n_in + 2 * khalf;
  v8f c = {0.f, 0.f, 0.f, 0.f, 0.f, 0.f, 0.f, 0.f};
  for (int k0 = 0; k0 < n_in; k0 += 4) {
    v2f a, b;
    a.x = arow[k0]; a.y = arow[k0 + 1];
    b.x = brow[k0]; b.y = brow[k0 + 1];
    c = __builtin_amdgcn_wmma_f32_16x16x4_f32(false, a, false, b, (short)0, c, false, false);
  }
  size_t obase = (size_t)(row0 + 8 * khalf) * n_out + (size_t)(col0 + mrow);
#pragma unroll
  for (int v = 0; v < 8; ++v)
    out[obase + (size_t)v * n_out] = c[v];
}

// fused variant: final upsample to W=64 accumulates into out and residual
__global__ void ms_resize_w_wmma_accum(const float* __restrict__ in,
                                       float* __restrict__ outAcc, float* __restrict__ resid,
                                       const float* __restrict__ Wt, int n_in, int n_out) {
  int wave = (blockIdx.x * blockDim.x + threadIdx.x) >> 5;
  int lane = threadIdx.x & 31;
  int tilesN = n_out >> 4;
  int tileM = wave / tilesN;
  int tileN = wave - tileM * tilesN;
  int row0 = tileM << 4;
  int col0 = tileN << 4;
  int mrow  = lane & 15;
  int khalf = lane >> 4;
  const float* arow = in + (size_t)(row0 + mrow) * n_in + 2 * khalf;
  const float* brow = Wt + (size_t)(col0 + mrow) * n_in + 2 * khalf;
  v8f c = {0.f, 0.f, 0.f, 0.f, 0.f, 0.f, 0.f, 0.f};
  for (int k0 = 0; k0 < n_in; k0 += 4) {
    v2f a, b;
    a.x = arow[k0]; a.y = arow[k0 + 1];
    b.x = brow[k0]; b.y = brow[k0 + 1];
    c = __builtin_amdgcn_wmma_f32_16x16x4_f32(false, a, false, b, (short)0, c, false, false);
  }
  size_t obase = (size_t)(row0 + 8 * khalf) * n_out + (size_t)(col0 + mrow);
#pragma unroll
  for (int v = 0; v < 8; ++v) {
    size_t idx = obase + (size_t)v * n_out;
    float val = c[v];
    outAcc[idx] += val;
    resid[idx]  -= val;
  }
}

// ---------------- BSQ quantize (wave per token): l2norm over C=64, sign code,
// entropy + commit + p-mean partial sums. In-place z -> q. --------------------
__global__ void ms_bsq_quantize(float* __restrict__ z, int ntok, int S,
                                float* __restrict__ red /* psum[64], pse, commit */) {
  __shared__ float lps[66];
  int tid = threadIdx.x;
  if (tid < 64) lps[tid] = 0.0f;
  if (tid == 0) { lps[64] = 0.0f; lps[65] = 0.0f; }
  __syncthreads();

  int wave = (blockIdx.x * blockDim.x + tid) >> 5;
  int lane = tid & 31;
  if (wave < ntok) {
    int b = wave / S;
    int s = wave - b * S;
    long base = (long)b * 64 * S + s;
    int c0 = lane, c1 = lane + 32;
    float z0 = z[base + (long)c0 * S];
    float z1 = z[base + (long)c1 * S];
    float ss = z0 * z0 + z1 * z1;
    for (int o = 16; o >= 1; o >>= 1) ss += __shfl_xor(ss, o, 32);
    float inv = 1.0f / fmaxf(sqrtf(ss), 1e-12f);
    float zn0 = z0 * inv, zn1 = z1 * inv;
    float q0 = zn0 > 0.0f ? 0.125f : -0.125f;   // 1/sqrt(64)
    float q1 = zn1 > 0.0f ? 0.125f : -0.125f;
    z[base + (long)c0 * S] = q0;
    z[base + (long)c1 * S] = q1;
    // p = sigmoid(-4 * zn / sqrt(d) * INV_TEMP) = sigmoid(-50 zn)
    float p0 = 1.0f / (1.0f + __expf(50.0f * zn0));
    float p1 = 1.0f / (1.0f + __expf(50.0f * zn1));
    float h0 = -(p0 * __logf(p0 + 1e-8f) + (1.0f - p0) * __logf(1.0f - p0 + 1e-8f));
    float h1 = -(p1 * __logf(p1 + 1e-8f) + (1.0f - p1) * __logf(1.0f - p1 + 1e-8f));
    float cm = (zn0 - q0) * (zn0 - q0) + (zn1 - q1) * (zn1 - q1);
    atomicAdd(&lps[c0], p0);
    atomicAdd(&lps[c1], p1);
    float hsum = h0 + h1, csum = cm;
    for (int o = 16; o >= 1; o >>= 1) {
      hsum += __shfl_xor(hsum, o, 32);
      csum += __shfl_xor(csum, o, 32);
    }
    if (lane == 0) { atomicAdd(&lps[64], hsum); atomicAdd(&lps[65], csum); }
  }
  __syncthreads();
  if (tid < 64) atomicAdd(&red[tid], lps[tid]);
  if (tid == 0) { atomicAdd(&red[64], lps[64]); atomicAdd(&red[65], lps[65]); }
}

// ---------------- aux finisher ------------------------------------------------
__global__ void ms_finish_aux(const float* __restrict__ red, float* __restrict__ out_aux) {
  if (blockIdx.x != 0 || threadIdx.x != 0) return;
  const int tt[8] = {1, 1, 1, 2, 3, 5, 5, 5};
  const int hh[8] = {1, 2, 4, 8, 16, 32, 48, 64};
  float total = 0.0f;
  for (int s = 0; s < 8; ++s) {
    float ntok = 8.0f * (float)(tt[s] * hh[s] * hh[s]);
    const float* r = red + s * 68;
    float cb = 0.0f;
    for (int d = 0; d < 64; ++d) {
      float pm = r[d] / ntok;
      cb += -(pm * __logf(pm + 1e-8f) + (1.0f - pm) * __logf(1.0f - pm + 1e-8f));
    }
    float pse    = r[64] / ntok;
    float commit = r[65] / (ntok * 64.0f);
    // aux = ENTROPY_W*(GAMMA0*pse - GAMMA*cb) + COMMIT_W*commit
    total += 0.1f * (pse - cb) + 0.25f * commit;
  }
  out_aux[0] = total / 8.0f;
}

// ---------------- host orchestration -----------------------------------------
extern "C" void kernel_launch(void* const* d_in, const int* in_sizes, int n_in,
                              void* d_out, int out_size, void* d_ws, size_t ws_size,
                              hipStream_t stream) {
  const float* x = (const float*)d_in[0];
  float* out = (float*)d_out;             // [0,NELEM): quantized_out; [NELEM]: aux
  float* ws  = (float*)d_ws;

  static const int st[8] = {1, 1, 1, 2, 3, 5, 5, 5};
  static const int sh[8] = {1, 2, 4, 8, 16, 32, 48, 64};

  // workspace layout (float offsets)
  float* red = ws;                         // 8 * 68 floats of reduction slots
  size_t off = 1024;
  size_t wDT[8], wDH[8], wDW[8], wUT[8], wUH[8], wUW[8];
  for (int s = 0; s < 8; ++s) {
    wDT[s] = off; off += (size_t)st[s] * 5;
    wDH[s] = off; off += (size_t)sh[s] * 64;
    wDW[s] = off; off += (size_t)sh[s] * 64;
    wUT[s] = off; off += (size_t)5 * st[s];
    wUH[s] = off; off += (size_t)64 * sh[s];
    wUW[s] = off; off += (size_t)64 * sh[s];
  }
  float* R    = ws + 65536;                // residual, NELEM floats
  float* tmpA = R + NELEM;
  float* tmpB = tmpA + NELEM;
  (void)ws_size; (void)in_sizes; (void)n_in; (void)out_size;

  ms_init<<<(NELEM + 255) / 256, 256, 0, stream>>>(x, R, out, red, NELEM, 8 * 68);

  for (int s = 0; s < 8; ++s) {
    ms_weights<<<1, 64, 0, stream>>>(ws + wDT[s], 5,     st[s]);
    ms_weights<<<1, 64, 0, stream>>>(ws + wDH[s], 64,    sh[s]);
    ms_weights<<<1, 64, 0, stream>>>(ws + wDW[s], 64,    sh[s]);
    ms_weights<<<1, 64, 0, stream>>>(ws + wUT[s], st[s], 5);
    ms_weights<<<1, 64, 0, stream>>>(ws + wUH[s], sh[s], 64);
    ms_weights<<<1, 64, 0, stream>>>(ws + wUW[s], sh[s], 64);
  }

  for (int s = 0; s < 8; ++s) {
    const int t = st[s], h = sh[s], w = sh[s];
    const int S = t * h * w;

    // ---- downsample: W then H then T (R -> A -> B -> A) ----
    if ((w & 15) == 0) {
      int waves = (ROWS_W / 16) * (w / 16);
      ms_resize_w_wmma<<<waves * 32 / 256, 256, 0, stream>>>(R, tmpA, ws + wDW[s], 64, w);
    } else {
      int tot = ROWS_W * w;
      ms_resize_axis<<<(tot + 255) / 256, 256, 0, stream>>>(R, tmpA, ws + wDW[s],
                                                            ROWS_W, 64, w, 1);
    }
    {
      int tot = BB * CC * TDIM * h * w;   // outer=2560, 64 -> h, inner=w
      ms_resize_axis<<<(tot + 255) / 256, 256, 0, stream>>>(tmpA, tmpB, ws + wDH[s],
                                                            BB * CC * TDIM, 64, h, w);
    }
    {
      int tot = BB * CC * t * h * w;      // outer=512, 5 -> t, inner=h*w
      ms_resize_axis<<<(tot + 255) / 256, 256, 0, stream>>>(tmpB, tmpA, ws + wDT[s],
                                                            BB * CC, 5, t, h * w);
    }

    // ---- quantize in place (z -> q) + statistics ----
    {
      int ntok = BB * S;
      int blocks = (ntok + 3) / 4;        // 128 threads = 4 waves, wave per token
      ms_bsq_quantize<<<blocks, 128, 0, stream>>>(tmpA, ntok, S, red + s * 68);
    }

    // ---- upsample: T then H then W-fused (A -> B -> A -> (out,R)) ----
    {
      int tot = BB * CC * TDIM * h * w;   // outer=512, t -> 5, inner=h*w
      ms_resize_axis<<<(tot + 255) / 256, 256, 0, stream>>>(tmpA, tmpB, ws + wUT[s],
                                                            BB * CC, t, 5, h * w);
    }
    {
      int tot = BB * CC * TDIM * 64 * w;  // outer=2560, h -> 64, inner=w
      ms_resize_axis<<<(tot + 255) / 256, 256, 0, stream>>>(tmpB, tmpA, ws + wUH[s],
                                                            BB * CC * TDIM, h, 64, w);
    }
    if ((w & 15) == 0) {
      int waves = (ROWS_W / 16) * (64 / 16);
      ms_resize_w_wmma_accum<<<waves * 32 / 256, 256, 0, stream>>>(tmpA, out, R,
                                                                   ws + wUW[s], w, 64);
    } else {
      ms_resize_axis_accum<<<(NELEM + 255) / 256, 256, 0, stream>>>(tmpA, ws + wUW[s],
                                                                    out, R,
                                                                    ROWS_W, w, 64);
    }
  }

  ms_finish_aux<<<1, 1, 0, stream>>>(red, out + NELEM);
}